// GCN_65085934404070
// MI455X (gfx1250) — compile-verified
//
#include <hip/hip_runtime.h>
#include <math.h>

typedef float v2f __attribute__((ext_vector_type(2)));
typedef float v8f __attribute__((ext_vector_type(8)));

#define IN_C 64
#define HID_C 128
#define OUT_C 16
#define N_GRAPHS 128

// ---------------------------------------------------------------- init
__global__ void k_init(float* __restrict__ A1, float* __restrict__ A2,
                       float* __restrict__ deg, float* __restrict__ pool,
                       float* __restrict__ cnt, long tot1, long tot2, int n) {
  long idx = (long)blockIdx.x * blockDim.x + threadIdx.x;
  if (idx < tot1) A1[idx] = 0.f;
  if (idx < tot2) A2[idx] = 0.f;
  if (idx < n)    deg[idx] = 1.0f;                 // self-loop contribution
  if (idx < N_GRAPHS * OUT_C) pool[idx] = 0.f;
  if (idx < N_GRAPHS)         cnt[idx]  = 0.f;
}

// ---------------------------------------------------------------- degree
__global__ void k_deg(const int* __restrict__ dst, float* __restrict__ deg, int nE) {
  int e = blockIdx.x * blockDim.x + threadIdx.x;
  if (e < nE) atomicAdd(&deg[dst[e]], 1.0f);
}

__global__ void k_dinv(const float* __restrict__ deg, float* __restrict__ dinv, int n) {
  int i = blockIdx.x * blockDim.x + threadIdx.x;
  if (i < n) {
    float d = deg[i];
    dinv[i] = d > 0.f ? rsqrtf(d) : 0.f;
  }
}

// ---------------------------------------------------------------- GEMM1: [n x 64] @ [64 x 128]
// block = 256 threads = 8 waves; block b -> rows [16b,16b+16); wave w -> cols [16w,16w+16)
__global__ void k_gemm1(const float* __restrict__ X, const float* __restrict__ W,
                        float* __restrict__ H, int n) {
  const int lane = threadIdx.x & 31;
  const int wv   = threadIdx.x >> 5;
  const int row0 = blockIdx.x * 16;
  const int col0 = wv * 16;
  const int m  = lane & 15;
  const int hi = lane >> 4;
  int row = row0 + m; if (row >= n) row = n - 1;       // clamp keeps EXEC all-1
  const float* xr = X + (size_t)row * IN_C;
  const float* wc = W + col0 + m;
  v8f acc = {};
#pragma unroll
  for (int k = 0; k < IN_C; k += 4) {
    const int ka = k + 2 * hi;
    v2f a, b;
    a.x = xr[ka];
    a.y = xr[ka + 1];
    b.x = wc[(size_t)ka * HID_C];
    b.y = wc[(size_t)(ka + 1) * HID_C];
    acc = __builtin_amdgcn_wmma_f32_16x16x4_f32(false, a, false, b, (short)0, acc,
                                                false, false);
  }
#pragma unroll
  for (int r = 0; r < 8; r++) {
    const int rm = row0 + r + 8 * hi;
    if (rm < n) H[(size_t)rm * HID_C + col0 + m] = acc[r];
  }
}

// ---------------------------------------------------------------- GEMM2: [n x 128] @ [128 x 16]
// block = 32 threads = 1 wave; block b -> rows [16b,16b+16), single 16-col tile
__global__ void k_gemm2(const float* __restrict__ Hr, const float* __restrict__ W,
                        float* __restrict__ H2, int n) {
  const int lane = threadIdx.x & 31;
  const int row0 = blockIdx.x * 16;
  const int m  = lane & 15;
  const int hi = lane >> 4;
  int row = row0 + m; if (row >= n) row = n - 1;
  const float* xr = Hr + (size_t)row * HID_C;
  const float* wc = W + m;
  v8f acc = {};
#pragma unroll
  for (int k = 0; k < HID_C; k += 4) {
    const int ka = k + 2 * hi;
    v2f a, b;
    a.x = xr[ka];
    a.y = xr[ka + 1];
    b.x = wc[ka * OUT_C];
    b.y = wc[(ka + 1) * OUT_C];
    acc = __builtin_amdgcn_wmma_f32_16x16x4_f32(false, a, false, b, (short)0, acc,
                                                false, false);
  }
#pragma unroll
  for (int r = 0; r < 8; r++) {
    const int rm = row0 + r + 8 * hi;
    if (rm < n) H2[(size_t)rm * OUT_C + m] = acc[r];
  }
}

// ---------------------------------------------------------------- edge scatter, layer 1
// one wave per edge: 32 lanes x float4 = 128 channels; coalesced 512B gather
__global__ void k_agg1(const int* __restrict__ src, const int* __restrict__ dst,
                       const float* __restrict__ dinv, const float* __restrict__ H1,
                       float* __restrict__ A1, int nE) {
  const int e = blockIdx.x * 8 + (threadIdx.x >> 5);
  if (e >= nE) return;                                 // wave-uniform
  const int lane = threadIdx.x & 31;
  const int s = src[e];
  const int d = dst[e];
  const float nrm = dinv[s] * dinv[d];
  const float4 v = ((const float4*)(H1 + (size_t)s * HID_C))[lane];
  float* out = A1 + (size_t)d * HID_C + lane * 4;
  atomicAdd(out + 0, v.x * nrm);
  atomicAdd(out + 1, v.y * nrm);
  atomicAdd(out + 2, v.z * nrm);
  atomicAdd(out + 3, v.w * nrm);
}

// ---------------------------------------------------------------- finalize layer1 (in place)
__global__ void k_fin1(const float* __restrict__ H1, const float* __restrict__ dinv,
                       const float* __restrict__ b1, float* __restrict__ A1, int n) {
  const int idx = blockIdx.x * blockDim.x + threadIdx.x;
  if (idx >= n * HID_C) return;
  const int i = idx >> 7;
  const int c = idx & (HID_C - 1);
  const float di = dinv[i];
  const float v = A1[idx] + H1[idx] * di * di + b1[c];   // self-loop + bias
  A1[idx] = v > 0.f ? v : 0.f;                            // ReLU
}

// ---------------------------------------------------------------- edge scatter, layer 2
// 4 lanes per edge, float4 per lane = 16 channels
__global__ void k_agg2(const int* __restrict__ src, const int* __restrict__ dst,
                       const float* __restrict__ dinv, const float* __restrict__ H2,
                       float* __restrict__ A2, int nE) {
  const int t = blockIdx.x * blockDim.x + threadIdx.x;
  const int e = t >> 2;
  if (e >= nE) return;
  const int q = (t & 3) * 4;
  const int s = src[e];
  const int d = dst[e];
  const float nrm = dinv[s] * dinv[d];
  const float4 v = *(const float4*)(H2 + (size_t)s * OUT_C + q);
  float* out = A2 + (size_t)d * OUT_C + q;
  atomicAdd(out + 0, v.x * nrm);
  atomicAdd(out + 1, v.y * nrm);
  atomicAdd(out + 2, v.z * nrm);
  atomicAdd(out + 3, v.w * nrm);
}

// ---------------------------------------------------------------- finalize layer2 + pooling
__global__ void k_fin2pool(const float* __restrict__ H2, const float* __restrict__ dinv,
                           const float* __restrict__ b2, const float* __restrict__ A2,
                           const int* __restrict__ batch, float* __restrict__ pool,
                           float* __restrict__ cnt, int n) {
  const int idx = blockIdx.x * blockDim.x + threadIdx.x;
  if (idx >= n * OUT_C) return;
  const int i = idx >> 4;
  const int c = idx & (OUT_C - 1);
  const float di = dinv[i];
  const float v = A2[idx] + H2[idx] * di * di + b2[c];
  const int g = batch[i];
  atomicAdd(&pool[g * OUT_C + c], v);
  if (c == 0) atomicAdd(&cnt[g], 1.0f);
}

// ---------------------------------------------------------------- mean + log_softmax
__global__ void k_softmax(const float* __restrict__ pool, const float* __restrict__ cnt,
                          float* __restrict__ out) {
  const int g = threadIdx.x;
  if (g >= N_GRAPHS) return;
  float c = cnt[g];
  if (c < 1.f) c = 1.f;
  float v[OUT_C];
  float mx = -1e30f;
#pragma unroll
  for (int j = 0; j < OUT_C; j++) {
    v[j] = pool[g * OUT_C + j] / c;
    mx = fmaxf(mx, v[j]);
  }
  float s = 0.f;
#pragma unroll
  for (int j = 0; j < OUT_C; j++) s += expf(v[j] - mx);
  const float lse = mx + logf(s);
#pragma unroll
  for (int j = 0; j < OUT_C; j++) out[g * OUT_C + j] = v[j] - lse;
}

// ================================================================ launcher
extern "C" void kernel_launch(void* const* d_in, const int* in_sizes, int n_in,
                              void* d_out, int out_size, void* d_ws, size_t ws_size,
                              hipStream_t stream) {
  const float* x     = (const float*)d_in[0];
  const int*   ei    = (const int*)d_in[1];
  const int*   batch = (const int*)d_in[2];
  const float* W1    = (const float*)d_in[3];
  const float* b1    = (const float*)d_in[4];
  const float* W2    = (const float*)d_in[5];
  const float* b2    = (const float*)d_in[6];

  const int n  = in_sizes[0] / IN_C;     // 100000
  const int nE = in_sizes[1] / 2;        // 1200000
  const int* src = ei;
  const int* dst = ei + nE;

  // workspace layout (floats)
  float* ws   = (float*)d_ws;
  float* H1   = ws;                              // n * 128
  float* A1   = H1   + (size_t)n * HID_C;        // n * 128 (becomes h_relu)
  float* H2   = A1   + (size_t)n * HID_C;        // n * 16
  float* A2   = H2   + (size_t)n * OUT_C;        // n * 16
  float* deg  = A2   + (size_t)n * OUT_C;        // n
  float* dinv = deg  + n;                        // n
  float* pool = dinv + n;                        // 128*16
  float* cnt  = pool + N_GRAPHS * OUT_C;         // 128

  const int nTiles = (n + 15) / 16;
  const long tot1 = (long)n * HID_C;
  const long tot2 = (long)n * OUT_C;

  k_init<<<(int)((tot1 + 255) / 256), 256, 0, stream>>>(A1, A2, deg, pool, cnt,
                                                        tot1, tot2, n);
  k_deg<<<(nE + 255) / 256, 256, 0, stream>>>(dst, deg, nE);
  k_dinv<<<(n + 255) / 256, 256, 0, stream>>>(deg, dinv, n);

  k_gemm1<<<nTiles, 256, 0, stream>>>(x, W1, H1, n);
  k_agg1<<<(nE + 7) / 8, 256, 0, stream>>>(src, dst, dinv, H1, A1, nE);
  k_fin1<<<(n * HID_C + 255) / 256, 256, 0, stream>>>(H1, dinv, b1, A1, n);

  k_gemm2<<<nTiles, 32, 0, stream>>>(A1, W2, H2, n);
  k_agg2<<<(nE * 4 + 255) / 256, 256, 0, stream>>>(src, dst, dinv, H2, A2, nE);
  k_fin2pool<<<(n * OUT_C + 255) / 256, 256, 0, stream>>>(H2, dinv, b2, A2, batch,
                                                          pool, cnt, n);

  k_softmax<<<1, 128, 0, stream>>>(pool, cnt, (float*)d_out);
}